// MultilabelCrossEntropyLoss_44676249813136
// MI455X (gfx1250) — compile-verified
//
#include <hip/hip_runtime.h>
#include <math.h>

// ---------------------------------------------------------------------------
// MultilabelCrossEntropyLoss for MI455X (gfx1250, wave32).
// Stage 1: one wave per row -> coalesced label load, divergent gather,
//          xor-tree wave reduction, -log(s+TOL), per-wave partial sums.
// Stage 2: single wave reduces all partials with V_WMMA_F32_16X16X4_F32
//          (exact f32 matrix-unit summation: D = A x ones + C).
// ---------------------------------------------------------------------------

#define TOLF 1e-6f

typedef float v2f __attribute__((ext_vector_type(2)));
typedef float v8f __attribute__((ext_vector_type(8)));

__global__ __launch_bounds__(256)
void mlce_partials_gfx1250(const float* __restrict__ prd,
                           const int*   __restrict__ tgt,
                           float*       __restrict__ partials,
                           int B, int L) {
    const int lane       = threadIdx.x & 31;
    const int waveInBlk  = threadIdx.x >> 5;
    const int wavesPerBlk = blockDim.x >> 5;
    const int w          = blockIdx.x * wavesPerBlk + waveInBlk;
    const int totalWaves = gridDim.x * wavesPerBlk;

    float acc = 0.0f;
    for (int b = w; b < B; b += totalWaves) {
        const size_t tbase = (size_t)b * 33;
        // lanes 0..31 load the 32 labels of this row in one coalesced request
        const int label = tgt[tbase + lane];
        const int nb    = tgt[tbase + 32];

        // prefetch next row's target data (global_prefetch_b8)
        const int b2 = b + totalWaves;
        if (b2 < B) {
            __builtin_prefetch(&tgt[(size_t)b2 * 33 + lane], 0, 1);
        }

        // masked gather: one divergent VMEM gather per row
        float v = 0.0f;
        if (lane < nb) {
            v = prd[(size_t)b * L + label];
        }

        // deterministic 32-lane xor-tree reduction (wave32)
        #pragma unroll
        for (int off = 16; off > 0; off >>= 1) {
            v += __shfl_xor(v, off, 32);
        }

        // all lanes hold s; accumulate loss (wave-uniform)
        acc += -logf(v + TOLF);
    }

    if (w < totalWaves && lane == 0) {
        partials[w] = acc;
    }
}

// Single-wave final reduction using the f32 WMMA path.
// n must be a multiple of 64. EXEC is all-1s in the WMMA loop (no divergence).
__global__ __launch_bounds__(32)
void mlce_reduce_wmma_gfx1250(const float* __restrict__ partials,
                              float*       __restrict__ out,
                              int n, float scale) {
    const int lane = threadIdx.x;   // 0..31, one wave

    v8f acc = {};                   // 16x16 f32 accumulator (8 VGPRs)
    v2f ones;
    ones.x = 1.0f;
    ones.y = 1.0f;                  // B = ones(4x16)

    for (int base = 0; base < n; base += 64) {
        v2f a;                      // A = 64 partials mapped into 16x4 (any bijection ok)
        a.x = partials[base + lane];
        a.y = partials[base + 32 + lane];
        // D[m][n] = sum_k A[m][k] + C[m][n]  -> exact f32 block row-sums
        acc = __builtin_amdgcn_wmma_f32_16x16x4_f32(
                  /*neg_a=*/false, a, /*neg_b=*/false, ones,
                  /*c_mod=*/(short)0, acc, /*reuse_a=*/false, /*reuse_b=*/false);
    }

    // Fold the 8 accumulator registers: lane L (<16) holds sum over M=0..7 of
    // column L; lane L+16 holds sum over M=8..15. Columns are identical, so
    // column 0's total == grand total.
    float t = ((acc[0] + acc[1]) + (acc[2] + acc[3])) +
              ((acc[4] + acc[5]) + (acc[6] + acc[7]));
    t += __shfl_xor(t, 16, 32);

    if (lane == 0) {
        out[0] = t * scale;
    }
}

extern "C" void kernel_launch(void* const* d_in, const int* in_sizes, int n_in,
                              void* d_out, int out_size, void* d_ws, size_t ws_size,
                              hipStream_t stream) {
    const float* prd = (const float*)d_in[0];
    const int*   tgt = (const int*)d_in[1];
    float*       out = (float*)d_out;

    const int B = in_sizes[1] / 33;           // tgt is (B, 33)
    const int L = (B > 0) ? (in_sizes[0] / B) : 0;

    // 512 blocks x 8 waves = 4096 partials (16 KB). Shrink if ws is tiny,
    // keeping totalWaves a power of two >= 64 (multiple of 64 for stage 2).
    int nBlocks = 512;
    while ((size_t)(nBlocks * 8) * sizeof(float) > ws_size && nBlocks > 8) {
        nBlocks >>= 1;
    }
    const int totalWaves = nBlocks * 8;       // multiple of 64
    float* partials = (float*)d_ws;

    mlce_partials_gfx1250<<<nBlocks, 256, 0, stream>>>(prd, tgt, partials, B, L);
    mlce_reduce_wmma_gfx1250<<<1, 32, 0, stream>>>(partials, out, totalWaves,
                                                   1.0f / (float)B);
}